// GNN_6141803233890
// MI455X (gfx1250) — compile-verified
//
#include <hip/hip_runtime.h>

typedef float     v2f  __attribute__((ext_vector_type(2)));
typedef float     v8f  __attribute__((ext_vector_type(8)));

#define THREADS 256

// ---------------- init: deg=1 (self-loop), agg1=0, out=0 ----------------
__global__ __launch_bounds__(THREADS) void k_init(float* __restrict__ deg,
                                                  float* __restrict__ agg1,
                                                  float* __restrict__ out, int N) {
  int j = blockIdx.x * blockDim.x + threadIdx.x;
  if (j < 8 * N) agg1[j] = 0.0f;
  if (j < N) { deg[j] = 1.0f; out[j] = 0.0f; }
}

// ---------------- degree: deg[dst] += 1 over real edges ----------------
__global__ __launch_bounds__(THREADS) void k_deg(const long long* __restrict__ dst,
                                                 float* __restrict__ deg, int E) {
  int e = blockIdx.x * blockDim.x + threadIdx.x;
  if (e < E) unsafeAtomicAdd(&deg[(int)dst[e]], 1.0f);
}

// ---------------- dis = rsqrt(deg) in place ----------------
__global__ __launch_bounds__(THREADS) void k_dis(float* __restrict__ deg, int N) {
  int i = blockIdx.x * blockDim.x + threadIdx.x;
  if (i < N) deg[i] = rsqrtf(deg[i]);
}

// ---------------- h1 = x @ W1 via WMMA (one 16-node tile per wave) ----------------
__global__ __launch_bounds__(THREADS) void k_gemm1(const float* __restrict__ x,
                                                   const float* __restrict__ W1,
                                                   float* __restrict__ h1, int N) {
  const int lane = threadIdx.x & 31;
  const int tile = blockIdx.x * (blockDim.x >> 5) + (threadIdx.x >> 5);
  const int base = tile * 16;
  if (base >= N) return;                 // wave-uniform exit, EXEC stays full
  const int m    = lane & 15;            // A row / B,D column for this lane
  const int h    = lane >> 4;            // lane-half selects K pair / D row half
  const int mc   = m & 7;                // clamped B column (unconditional loads)
  const bool bok = (m < 8);              // W1 has only 8 output columns
  int row = base + m; if (row >= N) row = N - 1;   // clamp (N%16==0 in practice)

  v8f acc = {0.f,0.f,0.f,0.f,0.f,0.f,0.f,0.f};

#if defined(__HIP_DEVICE_COMPILE__)
  // Exact fp32 path: 4 chained K=4 WMMAs cover F_IN=16.
  #pragma unroll
  for (int c = 0; c < 4; ++c) {
    const int k0 = 4 * c + 2 * h;        // A/B: VGPR0->K=k0, VGPR1->K=k0+1
    v2f a, b;
    a.x = x[(size_t)row * 16 + k0];
    a.y = x[(size_t)row * 16 + k0 + 1];
    // Unconditional loads from clamped index + v_cndmask zero-pad (no exec branches)
    const float w0 = W1[(k0    ) * 8 + mc];
    const float w1 = W1[(k0 + 1) * 8 + mc];
    b.x = bok ? w0 : 0.0f;
    b.y = bok ? w1 : 0.0f;
    acc = __builtin_amdgcn_wmma_f32_16x16x4_f32(false, a, false, b,
                                                (short)0, acc, false, false);
  }
#endif

  // D layout: VGPR v -> row base+v+8h, column m; keep only 8 output features.
  if (base + 16 <= N) {                  // wave-uniform full-tile fast path
    float* p = h1 + (size_t)(base + 8 * h) * 8 + m;  // + v*8 floats per VGPR
    if (bok) {
      #pragma unroll
      for (int v = 0; v < 8; ++v) p[v * 8] = acc[v];  // 8 stores, imm offsets
    }
  } else {                               // tail tile (not taken for N=100000)
    #pragma unroll
    for (int v = 0; v < 8; ++v) {
      const int r = base + v + 8 * h;
      if (bok && r < N) h1[(size_t)r * 8 + m] = acc[v];
    }
  }
}

// ---------------- layer-1 edge scatter: agg1[dst] += norm * h1[src] ----------------
__global__ __launch_bounds__(THREADS) void k_edge1(const long long* __restrict__ src,
                                                   const long long* __restrict__ dst,
                                                   const float* __restrict__ dis,
                                                   const float* __restrict__ h1,
                                                   float* __restrict__ agg1, int E) {
  int e = blockIdx.x * blockDim.x + threadIdx.x;
  if (e >= E) return;
  const int s = (int)src[e], d = (int)dst[e];
  const float w = dis[s] * dis[d];
  const float4* hp = reinterpret_cast<const float4*>(h1 + (size_t)s * 8);
  const float4 h0 = hp[0], h4 = hp[1];
  float* ap = agg1 + (size_t)d * 8;
  unsafeAtomicAdd(ap + 0, w * h0.x);
  unsafeAtomicAdd(ap + 1, w * h0.y);
  unsafeAtomicAdd(ap + 2, w * h0.z);
  unsafeAtomicAdd(ap + 3, w * h0.w);
  unsafeAtomicAdd(ap + 4, w * h4.x);
  unsafeAtomicAdd(ap + 5, w * h4.y);
  unsafeAtomicAdd(ap + 6, w * h4.z);
  unsafeAtomicAdd(ap + 7, w * h4.w);
}

// ------- self-loop + bias + ReLU, then fold layer-2 GEMM: z = relu(.) @ W2 -------
__global__ __launch_bounds__(THREADS) void k_self1(const float* __restrict__ dis,
                                                   const float* __restrict__ h1,
                                                   const float* __restrict__ agg1,
                                                   const float* __restrict__ b1,
                                                   const float* __restrict__ W2,
                                                   float* __restrict__ z, int N) {
  int i = blockIdx.x * blockDim.x + threadIdx.x;
  if (i >= N) return;
  const float w = dis[i] * dis[i];       // self-loop norm
  float acc = 0.0f;
  #pragma unroll
  for (int j = 0; j < 8; ++j) {
    float a = agg1[(size_t)i * 8 + j] + w * h1[(size_t)i * 8 + j] + b1[j];
    a = fmaxf(a, 0.0f);
    acc = fmaf(a, W2[j], acc);
  }
  z[i] = acc;
}

// ---------------- layer-2 edge scatter: out[dst] += norm * z[src] ----------------
__global__ __launch_bounds__(THREADS) void k_edge2(const long long* __restrict__ src,
                                                   const long long* __restrict__ dst,
                                                   const float* __restrict__ dis,
                                                   const float* __restrict__ z,
                                                   float* __restrict__ out, int E) {
  int e = blockIdx.x * blockDim.x + threadIdx.x;
  if (e >= E) return;
  const int s = (int)src[e], d = (int)dst[e];
  unsafeAtomicAdd(&out[d], dis[s] * dis[d] * z[s]);
}

// ---------------- self-loop term of layer 2 + b2 ----------------
__global__ __launch_bounds__(THREADS) void k_final(const float* __restrict__ dis,
                                                   const float* __restrict__ z,
                                                   const float* __restrict__ b2,
                                                   float* __restrict__ out, int N) {
  int i = blockIdx.x * blockDim.x + threadIdx.x;
  if (i < N) out[i] += dis[i] * dis[i] * z[i] + b2[0];
}

extern "C" void kernel_launch(void* const* d_in, const int* in_sizes, int n_in,
                              void* d_out, int out_size, void* d_ws, size_t ws_size,
                              hipStream_t stream) {
  const float*     x  = (const float*)d_in[0];
  const long long* ei = (const long long*)d_in[1];   // int64 edge_index [2, E]
  const float*     W1 = (const float*)d_in[2];       // [16, 8] row-major
  const float*     b1 = (const float*)d_in[3];       // [8]
  const float*     W2 = (const float*)d_in[4];       // [8]  (8x1)
  const float*     b2 = (const float*)d_in[5];       // [1]

  const int N = in_sizes[0] / 16;
  const int E = in_sizes[1] / 2;
  const long long* src = ei;
  const long long* dst = ei + E;
  float* out = (float*)d_out;

  // workspace layout (floats): dis[N] | h1[8N] | agg1[8N] | z[N]  (= 18N*4 B ~ 7.2 MB)
  float* dis  = (float*)d_ws;
  float* h1   = dis + N;
  float* agg1 = h1 + (size_t)8 * N;
  float* z    = agg1 + (size_t)8 * N;

  auto cdiv = [](int a, int b) { return (a + b - 1) / b; };

  k_init <<<cdiv(8 * N, THREADS), THREADS, 0, stream>>>(dis, agg1, out, N);
  k_deg  <<<cdiv(E, THREADS),     THREADS, 0, stream>>>(dst, dis, E);
  k_dis  <<<cdiv(N, THREADS),     THREADS, 0, stream>>>(dis, N);

  const int tiles = cdiv(N, 16);                      // one wave per 16-node tile
  k_gemm1<<<cdiv(tiles, THREADS / 32), THREADS, 0, stream>>>(x, W1, h1, N);

  k_edge1<<<cdiv(E, THREADS), THREADS, 0, stream>>>(src, dst, dis, h1, agg1, E);
  k_self1<<<cdiv(N, THREADS), THREADS, 0, stream>>>(dis, h1, agg1, b1, W2, z, N);
  k_edge2<<<cdiv(E, THREADS), THREADS, 0, stream>>>(src, dst, dis, z, out, E);
  k_final<<<cdiv(N, THREADS), THREADS, 0, stream>>>(dis, z, b2, out, N);
}